// NoseHooverChain_14491219657168
// MI455X (gfx1250) — compile-verified
//
#include <hip/hip_runtime.h>

// Nose-Hoover chain MD, B=1024 systems, NDOF=64, N_CHAIN=2, 200 steps.
// One wave32 per system; lane l holds DOF {2l, 2l+1} as float2.
// Per-step wave reduction done with V_WMMA_F32_16X16X4_F32 (A = ones) +
// 4 ds_swizzle XOR steps. Trajectory written with non-temporal b64 stores.

typedef float v2f __attribute__((ext_vector_type(2)));
typedef float v8f __attribute__((ext_vector_type(8)));

#define B_SYS    1024
#define NDOF_    64
#define DT_      0.002f
#define N_STEPS_ 200
#define WAVES_PER_BLOCK 8
#define THREADS_ (WAVES_PER_BLOCK * 32)

// Full-wave (32-lane) sum of the 64 values {a,b} per lane.
// Step 1: WMMA f32 16x16x4 with A = all-ones; B = our 64 values (2 VGPRs).
//         D row 0 (VGPR0) = 16 column sums, replicated lanes 0-15 / 16-31.
// Step 2: XOR-swizzle tree over 16 lanes -> total in every lane.
__device__ __forceinline__ float wave_sum64(float a, float b) {
  v2f ones; ones[0] = 1.0f; ones[1] = 1.0f;
  v2f bb;   bb[0] = a;      bb[1] = b;
  v8f c = {};
  v8f d = __builtin_amdgcn_wmma_f32_16x16x4_f32(
      /*neg_a=*/false, ones, /*neg_b=*/false, bb,
      /*c_mod=*/(short)0, c, /*reuse_a=*/false, /*reuse_b=*/false);
  float s = d[0];
  // group-of-32 ds_swizzle: offset = (xor_mask<<10) | (or=0<<5) | (and=0x1F)
  s += __int_as_float(__builtin_amdgcn_ds_swizzle(__float_as_int(s), (1 << 10) | 0x1F));
  s += __int_as_float(__builtin_amdgcn_ds_swizzle(__float_as_int(s), (2 << 10) | 0x1F));
  s += __int_as_float(__builtin_amdgcn_ds_swizzle(__float_as_int(s), (4 << 10) | 0x1F));
  s += __int_as_float(__builtin_amdgcn_ds_swizzle(__float_as_int(s), (8 << 10) | 0x1F));
  return s;  // every lane: sum of all 64 inputs
}

__global__ __launch_bounds__(THREADS_)
void nhc_traj_kernel(const float* __restrict__ x0, const float* __restrict__ v0,
                     const float* __restrict__ kTp, const float* __restrict__ massp,
                     const float* __restrict__ Qp, float* __restrict__ out) {
  const int lane = threadIdx.x & 31;
  const int sys  = blockIdx.x * WAVES_PER_BLOCK + (threadIdx.x >> 5);

  const float kT   = kTp[0];
  const float mass = massp[0];
  const float Q    = Qp[0];
  const float invm = 1.0f / mass;
  const float invQ = 1.0f / Q;
  const float dt2 = 0.5f * DT_, dt4 = 0.25f * DT_, dt8 = 0.125f * DT_;
  const float ndof_kT = (float)NDOF_ * kT;

  const size_t TRAJ  = (size_t)(N_STEPS_ + 1) * B_SYS * NDOF_;  // floats per traj array
  const size_t base  = (size_t)sys * NDOF_ + 2u * (unsigned)lane;

  v2f x = *(const v2f*)(x0 + base);
  v2f v = *(const v2f*)(v0 + base);

  // Frame 0: copies of x0, v0.
  __builtin_nontemporal_store(x, (v2f*)(out + base));
  __builtin_nontemporal_store(v, (v2f*)(out + TRAJ + base));

  // Live state: x, v (registers), thermostat xi0, xi1, carried KE.
  float KE  = 0.5f * mass * wave_sum64(v[0] * v[0], v[1] * v[1]);
  float xi0 = 0.0f, xi1 = 0.0f;

#pragma unroll 1
  for (int step = 0; step < N_STEPS_; ++step) {
    // ---- first half chain (j = 1 then j = 0) ----
    float G = (2.0f * KE - ndof_kT) * invQ;
    xi1 += dt4 * G;
    G = (Q * xi0 * xi0 - kT) * invQ;     // uses OLD xi0
    float s = __expf(-xi1 * dt8);
    xi0 = (xi0 * s + dt4 * G) * s;
    float c = __expf(-xi0 * dt2);        // same xi0 used for both v-scales
    v[0] *= c; v[1] *= c;
    // (post-scale KE/G in the reference are dead -> skipped)

    // ---- velocity-Verlet, F = -x ----
    v[0] -= dt2 * invm * x[0];  v[1] -= dt2 * invm * x[1];
    x[0] += DT_ * v[0];         x[1] += DT_ * v[1];
    v[0] -= dt2 * invm * x[0];  v[1] -= dt2 * invm * x[1];
    // (post-Verlet KE/G in the reference are dead -> skipped)

    // ---- second half chain (j = 0 then j = 1) ----
    v[0] *= c; v[1] *= c;                // exp(-xi0*dt/2), xi0 unchanged since first scale
    KE = 0.5f * mass * wave_sum64(v[0] * v[0], v[1] * v[1]);  // carried to next step
    G  = (2.0f * KE - ndof_kT) * invQ;
    s  = __expf(-xi1 * dt8);
    xi0 = (xi0 * s + dt4 * G) * s;
    G  = (Q * xi0 * xi0 - kT) * invQ;    // uses NEW xi0
    xi1 += dt4 * G;

    // ---- emit frame step+1 (coalesced 256B/wave, non-temporal) ----
    const size_t off = (size_t)(step + 1) * B_SYS * NDOF_ + base;
    __builtin_nontemporal_store(x, (v2f*)(out + off));
    __builtin_nontemporal_store(v, (v2f*)(out + TRAJ + off));
  }
}

extern "C" void kernel_launch(void* const* d_in, const int* in_sizes, int n_in,
                              void* d_out, int out_size, void* d_ws, size_t ws_size,
                              hipStream_t stream) {
  const float* x0   = (const float*)d_in[0];
  const float* v0   = (const float*)d_in[1];
  const float* kT   = (const float*)d_in[2];
  const float* mass = (const float*)d_in[3];
  const float* Q    = (const float*)d_in[4];
  float* out = (float*)d_out;

  dim3 grid(B_SYS / WAVES_PER_BLOCK);   // 128 blocks
  dim3 block(THREADS_);                 // 256 threads = 8 wave32 = 8 systems
  hipLaunchKernelGGL(nhc_traj_kernel, grid, block, 0, stream,
                     x0, v0, kT, mass, Q, out);
}